// ExpertClassifierBank_13056700580560
// MI455X (gfx1250) — compile-verified
//
#include <hip/hip_runtime.h>
#include <hip/hip_bf16.h>

// Problem constants (from reference setup_inputs)
#define B_SZ 4096
#define K_SZ 8
#define D_SZ 1024
#define C_SZ 100
#define CP   112                         // C padded to 7 tiles of 16
#define NT   7                           // N tiles per wave
#define WT_F2_PER_K ((D_SZ / 2) * CP)    // float2 elements per expert slot
#define WT_FLOATS   (K_SZ * WT_F2_PER_K * 2)

typedef __attribute__((ext_vector_type(2))) float v2f;
typedef __attribute__((ext_vector_type(8))) float v8f;

// ---------------------------------------------------------------------------
// Kernel 1: L2-normalize the K active experts' classifier rows, store them
// transposed + padded as interleaved d-pairs:
//   wt[k][p][c] = { w_norm[k][c][2p], w_norm[k][c][2p+1] }   (c < CP, p < D/2)
// so the GEMM's B-fragment load is one coalesced float2 per lane.
// Also computes scale[k] = min(exp(log_scales[idx[k]]), 100).
// Grid: K*CP blocks of 128 threads; each block owns one (k, c) row.
// ---------------------------------------------------------------------------
__global__ __launch_bounds__(128)
void prep_weights(const float* __restrict__ weights,
                  const int*   __restrict__ idx,
                  const float* __restrict__ log_scales,
                  float*       __restrict__ ws) {
  __shared__ float red[4];
  const int k   = blockIdx.x / CP;
  const int c   = blockIdx.x % CP;
  const int tid = threadIdx.x;
  float* wt_k = ws + (size_t)k * (WT_F2_PER_K * 2);

  if (c >= C_SZ) {  // zero the padding columns so WMMA on tile 6 is benign
    #pragma unroll
    for (int j = 0; j < 8; ++j) {
      int d = tid + j * 128;
      wt_k[(size_t)(d >> 1) * (2 * CP) + 2 * c + (d & 1)] = 0.0f;
    }
    return;
  }

  const int e = idx[k];
  const float* wrow = weights + ((size_t)e * C_SZ + c) * D_SZ;

  float vals[8];
  float ss = 0.0f;
  #pragma unroll
  for (int j = 0; j < 8; ++j) {
    float v = wrow[tid + j * 128];
    vals[j] = v;
    ss += v * v;
  }
  // wave32 butterfly reduction, then cross-wave via LDS
  #pragma unroll
  for (int off = 16; off > 0; off >>= 1) ss += __shfl_xor(ss, off, 32);
  if ((tid & 31) == 0) red[tid >> 5] = ss;
  __syncthreads();
  const float tot = red[0] + red[1] + red[2] + red[3];
  const float inv = 1.0f / fmaxf(sqrtf(tot), 1e-12f);

  #pragma unroll
  for (int j = 0; j < 8; ++j) {
    int d = tid + j * 128;
    wt_k[(size_t)(d >> 1) * (2 * CP) + 2 * c + (d & 1)] = vals[j] * inv;
  }
  if (c == 0 && tid == 0)
    ws[WT_FLOATS + k] = fminf(expf(log_scales[e]), 100.0f);
}

// ---------------------------------------------------------------------------
// Kernel 2: fp32 WMMA GEMM. One wave per (16-row M-tile, expert k).
// 7 accumulators cover all 112 (padded) output columns, so `pooled` is read
// exactly once. The x-row L2 norm
// is fused: each lane accumulates sum-of-squares of the A elements it loads.
// A fragment (16x4 f32): lane m=lane&15 is row m; g=lane>>4 selects K pair
//   {4s+2g, 4s+2g+1} -> one strided float2 load, 8-step L0 line reuse.
// B fragment (4x16 f32): lane n=lane&15 is column n, same K-pair split ->
//   one coalesced float2 load from the pre-transposed weights.
// ---------------------------------------------------------------------------
__global__ __launch_bounds__(32)
void cosine_gemm(const float* __restrict__ pooled,
                 const float* __restrict__ ws,
                 float*       __restrict__ out) {
  const int lane = threadIdx.x;
  const int m    = lane & 15;   // A row / B column owned by this lane
  const int g    = lane >> 4;   // K-pair group (0: k%4 in {0,1}, 1: {2,3})
  const int k    = blockIdx.y;
  const int row0 = blockIdx.x * 16;

  const float* aptr = pooled + ((size_t)(row0 + m) * K_SZ + k) * D_SZ + 2 * g;
  const v2f*   bptr = (const v2f*)ws + (size_t)k * WT_F2_PER_K
                                     + (size_t)g * CP + m;

  v8f acc[NT] = {};
  float ss = 0.0f;

  #pragma unroll 2
  for (int s = 0; s < D_SZ / 4; ++s) {
    v2f a = *(const v2f*)aptr;
    aptr += 4;
    ss += a.x * a.x + a.y * a.y;     // fused row-norm partial
    #pragma unroll
    for (int t = 0; t < NT; ++t) {
      v2f b = bptr[16 * t];
      acc[t] = __builtin_amdgcn_wmma_f32_16x16x4_f32(
          /*neg_a=*/false, a, /*neg_b=*/false, b,
          /*c_mod=*/(short)0, acc[t], /*reuse_a=*/false, /*reuse_b=*/false);
    }
    bptr += 2 * CP;
  }

  // lanes m and m+16 together hold row m's full sum of squares
  ss += __shfl_xor(ss, 16, 32);
  const float scale_k = ws[WT_FLOATS + k];

  // C/D layout: lane holds N=m; VGPR v holds M = 8*g + v  -> need 8 row factors
  float f[8];
  #pragma unroll
  for (int v = 0; v < 8; ++v) {
    float rss = __shfl(ss, 8 * g + v, 32);
    f[v] = scale_k / fmaxf(sqrtf(rss), 1e-12f);
  }

  #pragma unroll
  for (int t = 0; t < NT; ++t) {
    const int n = 16 * t + m;
    if (n < C_SZ) {
      #pragma unroll
      for (int v = 0; v < 8; ++v) {
        const int row = row0 + 8 * g + v;
        out[((size_t)row * K_SZ + k) * C_SZ + n] = acc[t][v] * f[v];
      }
    }
  }
}

// ---------------------------------------------------------------------------
extern "C" void kernel_launch(void* const* d_in, const int* in_sizes, int n_in,
                              void* d_out, int out_size, void* d_ws, size_t ws_size,
                              hipStream_t stream) {
  const float* pooled     = (const float*)d_in[0];
  const int*   idx        = (const int*)  d_in[1];
  const float* weights    = (const float*)d_in[2];
  const float* log_scales = (const float*)d_in[3];
  float* ws  = (float*)d_ws;   // [WT_FLOATS] normalized-transposed W, + K scales
  float* out = (float*)d_out;

  prep_weights<<<dim3(K_SZ * CP), dim3(128), 0, stream>>>(weights, idx,
                                                          log_scales, ws);
  cosine_gemm<<<dim3(B_SZ / 16, K_SZ), dim3(32), 0, stream>>>(pooled, ws, out);
}